// GCNNet_24498493456719
// MI455X (gfx1250) — compile-verified
//
#include <hip/hip_runtime.h>
#include <math.h>

// ---------------------------------------------------------------------------
// GCN 2-layer inference for MI455X (gfx1250, wave32, WMMA).
//   y = log_softmax( Anorm * relu(Anorm * (x W1) + b1) * W2 + b2 )
// Layer 1: transform-first (512 -> 16 before edges).
// Layer 2: aggregate-first (linearity commute: Anorm(h W2) == (Anorm h) W2),
//          cutting per-edge message width 40 -> 16 floats.
// Dense transforms use V_WMMA_F32_16X16X4_F32 (full fp32, tensor pipe).
// ---------------------------------------------------------------------------

typedef float v2f __attribute__((ext_vector_type(2)));
typedef float v8f __attribute__((ext_vector_type(8)));

#define FIN 512
#define HID 16
#define NCLS 40
#define NPAD 48   // 3 tiles of 16

// ---------------- degree / norm ----------------
__global__ void k_deg(const int* __restrict__ col, float* __restrict__ deg, int E) {
  int e = blockIdx.x * blockDim.x + threadIdx.x;
  if (e < E) atomicAdd(&deg[col[e]], 1.0f);
}

__global__ void k_dinv(float* __restrict__ d, int n) {
  int i = blockIdx.x * blockDim.x + threadIdx.x;
  if (i < n) d[i] = rsqrtf(d[i] + 1.0f);   // +1 = self loop; always > 0
}

// ---------------- WMMA GEMM 1: h1 = x @ W1  ([N,512]x[512,16]) ----------------
__global__ __launch_bounds__(256) void k_gemm1(const float* __restrict__ x,
                                               const float* __restrict__ W1,
                                               float* __restrict__ h1, int n) {
  __shared__ float sW[FIN * HID];           // 32 KB
  for (int i = threadIdx.x; i < FIN * HID; i += 256) sW[i] = W1[i];
  __syncthreads();

  const int wave = threadIdx.x >> 5;
  const int lane = threadIdx.x & 31;
  const int rowBase = (blockIdx.x * 8 + wave) * 16;
  const int m    = lane & 15;
  const int koff = (lane >> 4) << 1;        // 0 (lanes 0-15) or 2 (lanes 16-31)
  int row = rowBase + m;
  int rowc = row < n ? row : n - 1;         // clamp: keep EXEC all-1s for WMMA
  const float* xr = x + (size_t)rowc * FIN + koff;

  v8f c = {};
  for (int k = 0; k < FIN; k += 4) {
    v2f a = *(const v2f*)(xr + k);          // x[row][k+koff .. k+koff+1]
    v2f b;
    b.x = sW[(k + koff)     * HID + m];     // W1[k+koff  ][m]
    b.y = sW[(k + koff + 1) * HID + m];     // W1[k+koff+1][m]
    c = __builtin_amdgcn_wmma_f32_16x16x4_f32(false, a, false, b,
                                              (short)0, c, false, false);
  }
  // C/D layout: VGPR g -> row rowBase + g + 8*(lane>=16), col = m
  const int rb = rowBase + ((lane >> 4) << 3);
#pragma unroll
  for (int g = 0; g < 8; ++g) {
    int r = rb + g;
    if (r < n) h1[(size_t)r * HID + m] = c[g];
  }
}

// ---------------- edge scatter-add (16-wide messages) ----------------
__global__ void k_agg16(const int* __restrict__ rowi, const int* __restrict__ coli,
                        const float* __restrict__ dinv, const float* __restrict__ h,
                        float* __restrict__ agg, int E) {
  int e = blockIdx.x * blockDim.x + threadIdx.x;
  if (e >= E) return;
  int r = rowi[e], c = coli[e];
  float nrm = dinv[r] * dinv[c];
  const float4* hr = (const float4*)(h + (size_t)r * HID);
  float* ac = agg + (size_t)c * HID;
#pragma unroll
  for (int q = 0; q < 4; ++q) {
    float4 v = hr[q];
    atomicAdd(ac + q * 4 + 0, v.x * nrm);
    atomicAdd(ac + q * 4 + 1, v.y * nrm);
    atomicAdd(ac + q * 4 + 2, v.z * nrm);
    atomicAdd(ac + q * 4 + 3, v.w * nrm);
  }
}

// y1 = relu(agg1 + dinv^2 * h1 + b1), in place into agg1
__global__ void k_fin1(const float* __restrict__ h1, const float* __restrict__ b1,
                       const float* __restrict__ dinv, float* __restrict__ agg, int n) {
  int idx = blockIdx.x * blockDim.x + threadIdx.x;
  if (idx >= n * HID) return;
  int i = idx >> 4, f = idx & 15;
  float di = dinv[i];
  float v = agg[idx] + di * di * h1[idx] + b1[f];
  agg[idx] = fmaxf(v, 0.0f);
}

// z2 = agg2 + dinv^2 * y1, in place into agg2
__global__ void k_self2(const float* __restrict__ y1, const float* __restrict__ dinv,
                        float* __restrict__ agg2, int n) {
  int idx = blockIdx.x * blockDim.x + threadIdx.x;
  if (idx >= n * HID) return;
  int i = idx >> 4;
  float di = dinv[i];
  agg2[idx] += di * di * y1[idx];
}

// ---------------- WMMA GEMM 2: out = z2 @ W2 + b2  ([N,16]x[16,40]) ----------------
__global__ __launch_bounds__(256) void k_gemm2(const float* __restrict__ z2,
                                               const float* __restrict__ W2,
                                               const float* __restrict__ bias,
                                               float* __restrict__ out, int n) {
  __shared__ float sW[HID * NPAD];          // 16x48, zero padded
  __shared__ float sB[NPAD];
  for (int i = threadIdx.x; i < HID * NPAD; i += 256) {
    int kk = i / NPAD, cc = i % NPAD;
    sW[i] = (cc < NCLS) ? W2[kk * NCLS + cc] : 0.0f;
  }
  if (threadIdx.x < NPAD)
    sB[threadIdx.x] = (threadIdx.x < NCLS) ? bias[threadIdx.x] : 0.0f;
  __syncthreads();

  const int wave = threadIdx.x >> 5;
  const int lane = threadIdx.x & 31;
  const int rowBase = (blockIdx.x * 8 + wave) * 16;
  const int m    = lane & 15;
  const int koff = (lane >> 4) << 1;
  int row = rowBase + m;
  int rowc = row < n ? row : n - 1;
  const float* ar = z2 + (size_t)rowc * HID + koff;

  v8f c0 = {}, c1 = {}, c2 = {};
  for (int k = 0; k < HID; k += 4) {
    v2f a = *(const v2f*)(ar + k);
    v2f b0, b1, b2v;
    b0.x  = sW[(k + koff) * NPAD +  0 + m]; b0.y  = sW[(k + koff + 1) * NPAD +  0 + m];
    b1.x  = sW[(k + koff) * NPAD + 16 + m]; b1.y  = sW[(k + koff + 1) * NPAD + 16 + m];
    b2v.x = sW[(k + koff) * NPAD + 32 + m]; b2v.y = sW[(k + koff + 1) * NPAD + 32 + m];
    c0 = __builtin_amdgcn_wmma_f32_16x16x4_f32(false, a, false, b0,  (short)0, c0, false, false);
    c1 = __builtin_amdgcn_wmma_f32_16x16x4_f32(false, a, false, b1,  (short)0, c1, false, false);
    c2 = __builtin_amdgcn_wmma_f32_16x16x4_f32(false, a, false, b2v, (short)0, c2, false, false);
  }
  const int rb = rowBase + ((lane >> 4) << 3);
#pragma unroll
  for (int g = 0; g < 8; ++g) {
    int r = rb + g;
    if (r < n) {
      float* o = out + (size_t)r * NCLS;
      o[m]      = c0[g] + sB[m];
      o[16 + m] = c1[g] + sB[16 + m];
      if (m < 8) o[32 + m] = c2[g] + sB[32 + m];
    }
  }
}

// in-place row log_softmax over 40 classes
__global__ void k_logsoftmax(float* __restrict__ out, int n) {
  int i = blockIdx.x * blockDim.x + threadIdx.x;
  if (i >= n) return;
  float* r = out + (size_t)i * NCLS;
  float v[NCLS];
  float mx = -INFINITY;
#pragma unroll
  for (int c = 0; c < NCLS; ++c) { v[c] = r[c]; mx = fmaxf(mx, v[c]); }
  float s = 0.0f;
#pragma unroll
  for (int c = 0; c < NCLS; ++c) s += __expf(v[c] - mx);
  float ls = __logf(s);
#pragma unroll
  for (int c = 0; c < NCLS; ++c) r[c] = v[c] - mx - ls;
}

// ---------------------------------------------------------------------------
extern "C" void kernel_launch(void* const* d_in, const int* in_sizes, int n_in,
                              void* d_out, int out_size, void* d_ws, size_t ws_size,
                              hipStream_t stream) {
  const float* x   = (const float*)d_in[0];
  const int*   ei  = (const int*)d_in[1];       // [2,E] (bridge doc: ints -> int32)
  const float* W1  = (const float*)d_in[2];
  const float* b1  = (const float*)d_in[3];
  const float* W2  = (const float*)d_in[4];
  const float* b2  = (const float*)d_in[5];
  float*       out = (float*)d_out;

  const int N = in_sizes[0] / FIN;              // 100000
  const int E = in_sizes[1] / 2;                // 3200000
  const int* erow = ei;
  const int* ecol = ei + E;

  // workspace layout (floats): dinv[N] | h1[16N] | agg1[16N] | agg2[16N]
  float* ws    = (float*)d_ws;
  float* dinv  = ws;
  float* h1    = ws + (size_t)N;
  float* agg1  = ws + (size_t)N * 17;
  float* agg2  = ws + (size_t)N * 33;

  // zero deg and both accumulators (agg1,agg2 contiguous)
  hipMemsetAsync(dinv, 0, (size_t)N * sizeof(float), stream);
  hipMemsetAsync(agg1, 0, (size_t)N * 32 * sizeof(float), stream);

  const int TB = 256;
  const int gE   = (E + TB - 1) / TB;
  const int gN   = (N + TB - 1) / TB;
  const int gNH  = (N * HID + TB - 1) / TB;
  const int gTile = (N + 127) / 128;            // 8 waves x 16 rows per block

  k_deg   <<<gE,   TB, 0, stream>>>(ecol, dinv, E);
  k_dinv  <<<gN,   TB, 0, stream>>>(dinv, N);
  k_gemm1 <<<gTile,TB, 0, stream>>>(x, W1, h1, N);
  k_agg16 <<<gE,   TB, 0, stream>>>(erow, ecol, dinv, h1, agg1, E);
  k_fin1  <<<gNH,  TB, 0, stream>>>(h1, b1, dinv, agg1, N);        // agg1 -> y1
  k_agg16 <<<gE,   TB, 0, stream>>>(erow, ecol, dinv, agg1, agg2, E);
  k_self2 <<<gNH,  TB, 0, stream>>>(agg1, dinv, agg2, N);          // agg2 -> z2
  k_gemm2 <<<gTile,TB, 0, stream>>>(agg2, W2, b2, out, N);
  k_logsoftmax<<<gN, TB, 0, stream>>>(out, N);
}